// Vessel_Enhance_Block_31078383354238
// MI455X (gfx1250) — compile-verified
//
#include <hip/hip_runtime.h>
#include <cmath>

// Problem geometry
#define HH   512
#define WW   512
#define BATCH 16
#define IMGPIX (HH*WW)            // 262144
#define NPIX (BATCH*IMGPIX)       // 4194304
#define BIGLAB NPIX               // label sentinel for unmasked pixels
#define MAXLEN 20                 // max Gaussian kernel length (19) + pad
#define CC_ITERS 32

typedef float v2f __attribute__((ext_vector_type(2)));
typedef float v8f __attribute__((ext_vector_type(8)));

// Per-sigma Gaussian derivative weights, passed by value via kernarg.
struct SigmaW {
    float w0[MAXLEN];   // order 0
    float w1[MAXLEN];   // order 1
    float w2[MAXLEN];   // order 2
    float sigma2;
};

__device__ __forceinline__ int reflect512(int i) {
    // scipy 'reflect'/np 'symmetric' boundary: -1 -> 0, -2 -> 1, 512 -> 511 ...
    i = (i < 0) ? (-i - 1) : i;
    i = (i >= 512) ? (1023 - i) : i;
    return i;
}

// CDNA5 async global->LDS copy (ASYNCcnt-tracked, bypasses VGPR round-trip).
// LDS destination = per-lane VGPR byte offset; global addr = 64-bit per-lane VGPR pair.
__device__ __forceinline__ void async_f32_to_lds(unsigned lds_off, const float* gaddr) {
    asm volatile("global_load_async_to_lds_b32 %0, %1, off"
                 :: "v"(lds_off), "v"(gaddr)
                 : "memory");
}
__device__ __forceinline__ void wait_asynccnt0() {
    asm volatile("s_wait_asynccnt 0x0" ::: "memory");
}

// ---------------------------------------------------------------------------
// 1) scale + negate (Frangi black_ridges=False negates), zero the accumulator
// ---------------------------------------------------------------------------
__global__ void prep_kernel(const float* __restrict__ x,
                            float* __restrict__ neg,
                            float* __restrict__ acc) {
    int i = blockIdx.x * blockDim.x + threadIdx.x;
    if (i < NPIX) {
        neg[i] = -(x[i] * 500.0f);   // MIN_MAX = (0, 500)
        acc[i] = 0.0f;
    }
}

// ---------------------------------------------------------------------------
// 2a) Vertical separable conv for orders 0/1/2 via chained f32 WMMA 16x16x4.
//     One wave32 computes one 16x16 tile:  C(16x16) += A(16x4) x B(4x16)
//     A[m][u] = w[u-m] (banded weights), B[u][n] = img[y0-R+u][x0+n].
//     The 8 waves of a block share a (4*NCH)x128 input slab staged into LDS
//     with global_load_async_to_lds_b32 (reflect handled at staging time).
// ---------------------------------------------------------------------------
template<int R>
__global__ void __launch_bounds__(256)
vconv_wmma_kernel(const float* __restrict__ src,
                  float* __restrict__ v0, float* __restrict__ v1,
                  float* __restrict__ v2, SigmaW p) {
    constexpr int LEN  = 2 * R + 1;
    constexpr int KTOT = 16 + 2 * R;
    constexpr int NCH  = (KTOT + 3) / 4;
    constexpr int SLABROWS = 4 * NCH;          // multiple of 4, padded rows reflect

    __shared__ float slab[SLABROWS * 128];
    __shared__ float lw[3][MAXLEN];

    int t = threadIdx.x;
    if (t < MAXLEN) { lw[0][t] = p.w0[t]; lw[1][t] = p.w1[t]; lw[2][t] = p.w2[t]; }

    // Block decode: 128 blocks/image = 32 row-bands x 4 column-bands of 128.
    int bid = blockIdx.x;
    int img = bid >> 7;
    int rem = bid & 127;
    int y0  = (rem >> 2) << 4;
    int xb  = (rem & 3) << 7;
    const float* ip = src + (size_t)img * IMGPIX;

    // Async-stage the slab: SLABROWS x 128 f32, 256 threads -> 2 rows/pass.
    int colL = t & 127;
    int rowS = t >> 7;
    #pragma unroll
    for (int u = rowS; u < SLABROWS; u += 2) {
        int gy = reflect512(y0 - R + u);
        unsigned lofs = (unsigned)(unsigned long long)&slab[u * 128 + colL];
        async_f32_to_lds(lofs, ip + gy * WW + xb + colL);
    }
    wait_asynccnt0();
    __syncthreads();

    int lane = t & 31;
    int wv   = t >> 5;
    bool hi  = lane >= 16;
    int half = lane & 15;
    int x0l  = wv << 4;                        // wave's 16-col window in slab

    v8f c0 = {0.f,0.f,0.f,0.f,0.f,0.f,0.f,0.f};
    v8f c1 = c0, c2 = c0;

    #pragma unroll
    for (int ch = 0; ch < NCH; ++ch) {
        int kb = (ch << 2) + (hi ? 2 : 0);
        // B fragment from the LDS slab (lanes 0-15: 16 consecutive words/row)
        v2f b;
        b.x = slab[kb * 128 + x0l + half];
        b.y = slab[(kb + 1) * 128 + x0l + half];
        // A fragments: banded weight taps tap = u - m
        int t0 = kb - half, t1 = kb + 1 - half;
        bool g0 = (t0 >= 0) && (t0 < LEN);
        bool g1 = (t1 >= 0) && (t1 < LEN);
        int i0 = min(max(t0, 0), MAXLEN - 1);
        int i1 = min(max(t1, 0), MAXLEN - 1);
        v2f a0, a1, a2;
        a0.x = g0 ? lw[0][i0] : 0.0f;  a0.y = g1 ? lw[0][i1] : 0.0f;
        a1.x = g0 ? lw[1][i0] : 0.0f;  a1.y = g1 ? lw[1][i1] : 0.0f;
        a2.x = g0 ? lw[2][i0] : 0.0f;  a2.y = g1 ? lw[2][i1] : 0.0f;

        c0 = __builtin_amdgcn_wmma_f32_16x16x4_f32(false, a0, false, b, (short)0, c0, false, false);
        c1 = __builtin_amdgcn_wmma_f32_16x16x4_f32(false, a1, false, b, (short)0, c1, false, false);
        c2 = __builtin_amdgcn_wmma_f32_16x16x4_f32(false, a2, false, b, (short)0, c2, false, false);
    }

    // C/D layout: VGPR i -> row (y0 + (hi?8:0) + i), col x0+half
    int col   = xb + x0l + half;
    int rbase = y0 + (hi ? 8 : 0);
    size_t base = (size_t)img * IMGPIX + (size_t)rbase * WW + col;
    #pragma unroll
    for (int i = 0; i < 8; ++i) {
        v0[base + i * WW] = c0[i];
        v1[base + i * WW] = c1[i];
        v2[base + i * WW] = c2[i];
    }
}

// ---------------------------------------------------------------------------
// 2b) Horizontal conv (Hrr = k0*v2, Hrc = k1*v1, Hcc = k2*v0) fused with the
//     Hessian eigen / vesselness epilogue; acc = max(acc, vals).
//     A[m][u] = v[y0+m][x0-R+u], B[u][n] = w[u-n].
// ---------------------------------------------------------------------------
template<int R>
__global__ void __launch_bounds__(256)
hconv_frangi_kernel(const float* __restrict__ v0, const float* __restrict__ v1,
                    const float* __restrict__ v2, float* __restrict__ acc,
                    SigmaW p) {
    constexpr int LEN  = 2 * R + 1;
    constexpr int KTOT = 16 + 2 * R;
    constexpr int NCH  = (KTOT + 3) / 4;

    __shared__ float lw[3][MAXLEN];
    int t = threadIdx.x;
    if (t < MAXLEN) { lw[0][t] = p.w0[t]; lw[1][t] = p.w1[t]; lw[2][t] = p.w2[t]; }
    __syncthreads();

    int lane = t & 31;
    int wv   = t >> 5;
    int tile = blockIdx.x * 8 + wv;
    int img  = tile >> 10;
    int rem  = tile & 1023;
    int y0   = (rem >> 5) << 4;
    int x0   = (rem & 31) << 4;

    bool hi   = lane >= 16;
    int  half = lane & 15;

    v8f crr = {0.f,0.f,0.f,0.f,0.f,0.f,0.f,0.f};
    v8f crc = crr, ccc = crr;

    size_t rowoff = (size_t)img * IMGPIX + (size_t)(y0 + half) * WW;

    #pragma unroll
    for (int ch = 0; ch < NCH; ++ch) {
        int kb = (ch << 2) + (hi ? 2 : 0);
        // A fragments: row y0+half, cols x0-R+kb(+1), from v2/v1/v0
        int xa = reflect512(x0 - R + kb);
        int xc = reflect512(x0 - R + kb + 1);
        v2f aRR, aRC, aCC;
        aRR.x = v2[rowoff + xa];  aRR.y = v2[rowoff + xc];
        aRC.x = v1[rowoff + xa];  aRC.y = v1[rowoff + xc];
        aCC.x = v0[rowoff + xa];  aCC.y = v0[rowoff + xc];
        if (ch + 1 < NCH) {
            int xn = reflect512(x0 - R + kb + 4);
            __builtin_prefetch(&v2[rowoff + xn], 0, 1);   // global_prefetch_b8
        }
        // B fragments: banded weight taps tap = u - n
        int t0 = kb - half, t1 = kb + 1 - half;
        bool g0 = (t0 >= 0) && (t0 < LEN);
        bool g1 = (t1 >= 0) && (t1 < LEN);
        int i0 = min(max(t0, 0), MAXLEN - 1);
        int i1 = min(max(t1, 0), MAXLEN - 1);
        v2f b0, b1, b2;
        b0.x = g0 ? lw[0][i0] : 0.0f;  b0.y = g1 ? lw[0][i1] : 0.0f;
        b1.x = g0 ? lw[1][i0] : 0.0f;  b1.y = g1 ? lw[1][i1] : 0.0f;
        b2.x = g0 ? lw[2][i0] : 0.0f;  b2.y = g1 ? lw[2][i1] : 0.0f;

        crr = __builtin_amdgcn_wmma_f32_16x16x4_f32(false, aRR, false, b0, (short)0, crr, false, false);
        crc = __builtin_amdgcn_wmma_f32_16x16x4_f32(false, aRC, false, b1, (short)0, crc, false, false);
        ccc = __builtin_amdgcn_wmma_f32_16x16x4_f32(false, aCC, false, b2, (short)0, ccc, false, false);
    }

    // Fused vesselness epilogue
    int col   = x0 + half;
    int rbase = y0 + (hi ? 8 : 0);
    float s2  = p.sigma2;
    #pragma unroll
    for (int i = 0; i < 8; ++i) {
        size_t idx = (size_t)img * IMGPIX + (size_t)(rbase + i) * WW + col;
        float hrr = crr[i] * s2, hrc = crc[i] * s2, hcc = ccc[i] * s2;
        float mid  = 0.5f * (hrr + hcc);
        float dh   = 0.5f * (hrr - hcc);
        float disc = sqrtf(dh * dh + hrc * hrc);
        float ehi = mid + disc, elo = mid - disc;
        bool  small = fabsf(ehi) <= fabsf(elo);
        float lam1 = small ? ehi : elo;
        float lam2 = small ? elo : ehi;
        float lam2c = fmaxf(lam2, 1e-10f);
        float rr  = lam1 / lam2c;
        float rb2 = rr * rr;
        float ssq = lam1 * lam1 + lam2 * lam2;
        // BETA=1 -> exp(-rb2/2); GAMMA=5 -> 1-exp(-ssq/50)
        float val = expf(-rb2 * 0.5f) * (1.0f - expf(-ssq * (1.0f / 50.0f)));
        acc[idx] = fmaxf(acc[idx], val);
    }
}

// ---------------------------------------------------------------------------
// 3) Connected components (8-connected), remove components < 200 px.
// ---------------------------------------------------------------------------
__global__ void init_lab_kernel(const float* __restrict__ x,
                                const float* __restrict__ acc,
                                int* __restrict__ lab) {
    int i = blockIdx.x * blockDim.x + threadIdx.x;
    if (i >= NPIX) return;
    float v = x[i];
    bool m = (v >= 0.1f) && (v <= 0.5f) && (acc[i] >= 0.2f);
    lab[i] = m ? i : BIGLAB;
}

__global__ void nmin_kernel(const int* __restrict__ in, int* __restrict__ out) {
    int p = blockIdx.x * blockDim.x + threadIdx.x;
    if (p >= NPIX) return;
    int lp = in[p];
    if (lp == BIGLAB) { out[p] = BIGLAB; return; }
    int local = p & (IMGPIX - 1);
    int y = local >> 9, x = local & 511;
    int m = lp;
    #pragma unroll
    for (int dy = -1; dy <= 1; ++dy)
        #pragma unroll
        for (int dx = -1; dx <= 1; ++dx) {
            if (dy == 0 && dx == 0) continue;
            int yy = y + dy, xx = x + dx;
            if (yy >= 0 && yy < HH && xx >= 0 && xx < WW)
                m = min(m, in[p + dy * WW + dx]);
        }
    out[p] = m;
}

__global__ void jump_kernel(const int* __restrict__ in, int* __restrict__ out) {
    int p = blockIdx.x * blockDim.x + threadIdx.x;
    if (p >= NPIX) return;
    int v = in[p];
    if (v < BIGLAB) v = in[v];   // jump
    if (v < BIGLAB) v = in[v];   // jump again (pointer doubling)
    out[p] = v;
}

__global__ void zero_sizes_kernel(int* __restrict__ s) {
    int i = blockIdx.x * blockDim.x + threadIdx.x;
    if (i < NPIX) s[i] = 0;
}

__global__ void count_kernel(const int* __restrict__ lab, int* __restrict__ s) {
    int p = blockIdx.x * blockDim.x + threadIdx.x;
    if (p >= NPIX) return;
    int l = lab[p];
    if (l < BIGLAB) atomicAdd(&s[l], 1);
}

__global__ void final_kernel(const int* __restrict__ lab, const int* __restrict__ s,
                             float* __restrict__ out) {
    int p = blockIdx.x * blockDim.x + threadIdx.x;
    if (p >= NPIX) return;
    int l = lab[p];
    out[p] = (l < BIGLAB && s[l] >= 200) ? 1.0f : 0.0f;
}

// ---------------------------------------------------------------------------
// Host: scipy.ndimage-style Gaussian derivative kernel (truncate=4)
// ---------------------------------------------------------------------------
static void gauss1d(double sigma, int order, float* out) {
    int r = (int)(4.0 * sigma + 0.5);
    int len = 2 * r + 1;
    double phi[MAXLEN], s = 0.0;
    for (int i = 0; i < len; ++i) {
        double xv = (double)(i - r);
        phi[i] = std::exp(-0.5 * xv * xv / (sigma * sigma));
        s += phi[i];
    }
    for (int i = 0; i < MAXLEN; ++i) out[i] = 0.0f;
    double s2 = sigma * sigma;
    for (int i = 0; i < len; ++i) {
        double xv = (double)(i - r);
        double q = 1.0;
        if (order == 1) q = -xv / s2;
        else if (order == 2) q = xv * xv / (s2 * s2) - 1.0 / s2;
        out[i] = (float)(q * phi[i] / s);
    }
}

template<int R>
static void launch_sigma(const float* neg, float* v0, float* v1, float* v2,
                         float* acc, const SigmaW& p, hipStream_t stream) {
    const int BLOCKS = BATCH * 128;   // 128 blocks/image, 8 waves/block
    vconv_wmma_kernel<R><<<BLOCKS, 256, 0, stream>>>(neg, v0, v1, v2, p);
    hconv_frangi_kernel<R><<<BLOCKS, 256, 0, stream>>>(v0, v1, v2, acc, p);
}

extern "C" void kernel_launch(void* const* d_in, const int* in_sizes, int n_in,
                              void* d_out, int out_size, void* d_ws, size_t ws_size,
                              hipStream_t stream) {
    (void)in_sizes; (void)n_in; (void)out_size; (void)ws_size;
    const float* x = (const float*)d_in[0];
    float* out = (float*)d_out;
    char* ws = (char*)d_ws;

    const size_t SEG = (size_t)NPIX * sizeof(float);   // 16 MB segments
    float* neg = (float*)(ws);
    float* acc = (float*)(ws + 1 * SEG);
    float* v0  = (float*)(ws + 2 * SEG);
    float* v1  = (float*)(ws + 3 * SEG);
    float* v2  = (float*)(ws + 4 * SEG);
    // Aliased after the Frangi stage (conv temporaries are dead then):
    int* labA  = (int*)v0;
    int* labB  = (int*)v1;
    int* sizes = (int*)v2;

    dim3 blk(256);
    dim3 grdN((NPIX + 255) / 256);

    prep_kernel<<<grdN, blk, 0, stream>>>(x, neg, acc);

    const double sig[5] = {1.0, 1.3, 1.6, 1.9, 2.2};   // np.arange(1.0, 2.5, 0.3)
    for (int s = 0; s < 5; ++s) {
        SigmaW p{};
        gauss1d(sig[s], 0, p.w0);
        gauss1d(sig[s], 1, p.w1);
        gauss1d(sig[s], 2, p.w2);
        p.sigma2 = (float)(sig[s] * sig[s]);
        int radius = (int)(4.0 * sig[s] + 0.5);
        switch (radius) {
            case 4: launch_sigma<4>(neg, v0, v1, v2, acc, p, stream); break;
            case 5: launch_sigma<5>(neg, v0, v1, v2, acc, p, stream); break;
            case 6: launch_sigma<6>(neg, v0, v1, v2, acc, p, stream); break;
            case 8: launch_sigma<8>(neg, v0, v1, v2, acc, p, stream); break;
            default: launch_sigma<9>(neg, v0, v1, v2, acc, p, stream); break;
        }
    }

    init_lab_kernel<<<grdN, blk, 0, stream>>>(x, acc, labA);
    for (int it = 0; it < CC_ITERS; ++it) {
        nmin_kernel<<<grdN, blk, 0, stream>>>(labA, labB);
        jump_kernel<<<grdN, blk, 0, stream>>>(labB, labA);
    }
    zero_sizes_kernel<<<grdN, blk, 0, stream>>>(sizes);
    count_kernel<<<grdN, blk, 0, stream>>>(labA, sizes);
    final_kernel<<<grdN, blk, 0, stream>>>(labA, sizes, out);
}